// GCN_43568148251053
// MI455X (gfx1250) — compile-verified
//
#include <hip/hip_runtime.h>

#define N_NODES 50000
#define N_EDGES 1600000
#define IN_C    128
#define HID_C   64
#define OUT_C   40

typedef __attribute__((ext_vector_type(2))) float v2f;
typedef __attribute__((ext_vector_type(8))) float v8f;

// ---------------- degree / normalization ----------------

__global__ void k_init_deg(float* __restrict__ deg) {
    int i = blockIdx.x * blockDim.x + threadIdx.x;
    if (i < N_NODES) deg[i] = 1.0f;   // self-loop
}

__global__ void k_accum_deg(const long long* __restrict__ dst, float* __restrict__ deg) {
    int e = blockIdx.x * blockDim.x + threadIdx.x;
    if (e < N_EDGES) unsafeAtomicAdd(&deg[(int)dst[e]], 1.0f);
}

__global__ void k_rsqrt_deg(float* __restrict__ deg) {
    int i = blockIdx.x * blockDim.x + threadIdx.x;
    if (i < N_NODES) deg[i] = rsqrtf(deg[i]);
}

// ---------------- GEMM1: xw1 = x @ W1  (128 -> 64), WMMA f32 16x16x4 ----------------
// grid = 3125 blocks (16 rows each), block = 128 threads = 4 waves (one per N-tile of 16)

__global__ __launch_bounds__(128) void k_gemm1(const float* __restrict__ x,
                                               const float* __restrict__ W1,
                                               float* __restrict__ xw1) {
    __shared__ float lW[IN_C * HID_C];   // 32 KB, well within 320 KB/WGP
    const int tid = threadIdx.x;
    for (int i = tid; i < IN_C * HID_C; i += 128) lW[i] = W1[i];
    __syncthreads();

    const int wave  = tid >> 5;          // 0..3 = N-tile
    const int lane  = tid & 31;
    const int ln    = lane & 15;         // m for A, n for B/C
    const int khalf = lane >> 4;         // which K-pair this lane holds
    const int rb    = blockIdx.x * 16;   // 50000 % 16 == 0, no guards

    const float* arow = x + (size_t)(rb + ln) * IN_C + khalf * 2;
    const int    ncol = wave * 16 + ln;

    v8f c = {};
    #pragma unroll 4
    for (int kk = 0; kk < IN_C / 4; ++kk) {
        v2f a = *(const v2f*)(arow + kk * 4);           // A[m][k0+2h .. +1]
        v2f b;
        const int k0 = kk * 4 + khalf * 2;
        b.x = lW[(k0 + 0) * HID_C + ncol];              // B[k][n]
        b.y = lW[(k0 + 1) * HID_C + ncol];
        c = __builtin_amdgcn_wmma_f32_16x16x4_f32(false, a, false, b,
                                                  (short)0, c, false, false);
    }

    // C/D layout: vgpr r -> M = r + 8*(lane>=16), N = lane&15
    float* out = xw1 + (size_t)rb * HID_C + ncol;
    #pragma unroll
    for (int r = 0; r < 8; ++r)
        out[(size_t)(r + 8 * khalf) * HID_C] = c[r];
}

// ---------------- GEMM2: xw2 = h @ W2  (64 -> 40), WMMA f32 16x16x4 ----------------
// grid = 3125 blocks, block = 96 threads = 3 waves covering N = 48 (guard to 40)

__global__ __launch_bounds__(96) void k_gemm2(const float* __restrict__ h,
                                              const float* __restrict__ W2,
                                              float* __restrict__ xw2) {
    const int tid   = threadIdx.x;
    const int wave  = tid >> 5;          // 0..2 = N-tile
    const int lane  = tid & 31;
    const int ln    = lane & 15;
    const int khalf = lane >> 4;
    const int rb    = blockIdx.x * 16;

    const float* arow  = h + (size_t)(rb + ln) * HID_C + khalf * 2;
    const int    nglob = wave * 16 + ln;
    const bool   nok   = (nglob < OUT_C);

    v8f c = {};
    #pragma unroll
    for (int kk = 0; kk < HID_C / 4; ++kk) {
        v2f a = *(const v2f*)(arow + kk * 4);
        v2f b;
        const int k0 = kk * 4 + khalf * 2;
        b.x = nok ? W2[(k0 + 0) * OUT_C + nglob] : 0.0f;   // selects, EXEC stays all-1
        b.y = nok ? W2[(k0 + 1) * OUT_C + nglob] : 0.0f;
        c = __builtin_amdgcn_wmma_f32_16x16x4_f32(false, a, false, b,
                                                  (short)0, c, false, false);
    }

    if (nok) {
        #pragma unroll
        for (int r = 0; r < 8; ++r)
            xw2[(size_t)(rb + r + 8 * khalf) * OUT_C + nglob] = c[r];
    }
}

// ---------------- accumulator init: acc = xw * dis^2 + bias ----------------

__global__ void k_selfloop_bias(const float* __restrict__ xw,
                                const float* __restrict__ dis,
                                const float* __restrict__ bias,
                                float* __restrict__ acc, int C) {
    long long idx = (long long)blockIdx.x * blockDim.x + threadIdx.x;
    if (idx < (long long)N_NODES * C) {
        int i = (int)(idx / C), ch = (int)(idx % C);
        float d = dis[i];
        acc[idx] = xw[idx] * d * d + bias[ch];
    }
}

// ---------------- edge scatter: acc[dst] += xw[src] * dis[src]*dis[dst] ----------------
// 16 threads per edge, 4 channels each (float4 gather, 4 f32 atomics)

__global__ __launch_bounds__(256) void k_edge_agg(const float* __restrict__ xw,
                                                  const float* __restrict__ dis,
                                                  const long long* __restrict__ src,
                                                  const long long* __restrict__ dst,
                                                  float* __restrict__ acc, int C) {
    const int t    = threadIdx.x & 15;        // channel group
    const int eloc = threadIdx.x >> 4;        // edge within block
    const long long e = (long long)blockIdx.x * 16 + eloc;
    if (e >= N_EDGES) return;
    const int ch = t * 4;
    if (ch >= C) return;

    const int s = (int)src[e];
    const int d = (int)dst[e];
    const float norm = dis[s] * dis[d];

    const float4 v = *(const float4*)(xw + (size_t)s * C + ch);
    float* p = acc + (size_t)d * C + ch;
    unsafeAtomicAdd(p + 0, v.x * norm);
    unsafeAtomicAdd(p + 1, v.y * norm);
    unsafeAtomicAdd(p + 2, v.z * norm);
    unsafeAtomicAdd(p + 3, v.w * norm);
}

__global__ void k_relu(float* __restrict__ h, long long n) {
    long long i = (long long)blockIdx.x * blockDim.x + threadIdx.x;
    if (i < n) h[i] = fmaxf(h[i], 0.0f);
}

// ---------------- launcher ----------------

extern "C" void kernel_launch(void* const* d_in, const int* in_sizes, int n_in,
                              void* d_out, int out_size, void* d_ws, size_t ws_size,
                              hipStream_t stream) {
    (void)in_sizes; (void)n_in; (void)out_size; (void)ws_size;

    const float*     x   = (const float*)d_in[0];
    const long long* ei  = (const long long*)d_in[1];   // int64 [2, E]
    const float*     W1  = (const float*)d_in[2];
    const float*     b1  = (const float*)d_in[3];
    const float*     W2  = (const float*)d_in[4];
    const float*     b2  = (const float*)d_in[5];
    const long long* src = ei;
    const long long* dst = ei + N_EDGES;

    float* dis = (float*)d_ws;                          // N
    float* xw1 = dis + N_NODES;                         // N * 64
    float* h1  = xw1 + (size_t)N_NODES * HID_C;         // N * 64 (acc, then relu'd)
    float* xw2 = h1  + (size_t)N_NODES * HID_C;         // N * 40
    float* out = (float*)d_out;                         // N * 40 (acc in place)

    // normalization: deg -> rsqrt
    k_init_deg <<<(N_NODES + 255) / 256, 256, 0, stream>>>(dis);
    k_accum_deg<<<(N_EDGES + 255) / 256, 256, 0, stream>>>(dst, dis);
    k_rsqrt_deg<<<(N_NODES + 255) / 256, 256, 0, stream>>>(dis);

    // layer 1
    k_gemm1<<<N_NODES / 16, 128, 0, stream>>>(x, W1, xw1);
    {
        long long n = (long long)N_NODES * HID_C;
        k_selfloop_bias<<<(unsigned)((n + 255) / 256), 256, 0, stream>>>(xw1, dis, b1, h1, HID_C);
    }
    k_edge_agg<<<(N_EDGES + 15) / 16, 256, 0, stream>>>(xw1, dis, src, dst, h1, HID_C);
    {
        long long n = (long long)N_NODES * HID_C;
        k_relu<<<(unsigned)((n + 255) / 256), 256, 0, stream>>>(h1, n);
    }

    // layer 2
    k_gemm2<<<N_NODES / 16, 96, 0, stream>>>(h1, W2, xw2);
    {
        long long n = (long long)N_NODES * OUT_C;
        k_selfloop_bias<<<(unsigned)((n + 255) / 256), 256, 0, stream>>>(xw2, dis, b2, out, OUT_C);
    }
    k_edge_agg<<<(N_EDGES + 15) / 16, 256, 0, stream>>>(xw2, dis, src, dst, out, OUT_C);
}